// MultiHeadSelfAttentionKV_26955214749730
// MI455X (gfx1250) — compile-verified
//
#include <hip/hip_runtime.h>
#include <hip/hip_bf16.h>
#include <math.h>

// Problem constants (from reference): B=1
#define Lq 4096
#define Dm 512
#define Hh 8
#define Mm 2048
#define DH 64
#define KVN (Mm + Lq)   // 6144

typedef __attribute__((ext_vector_type(2))) float v2f;
typedef __attribute__((ext_vector_type(8))) float v8f;

static __device__ __forceinline__ v8f wmma_f32(v2f a, v2f b, v8f c) {
    // V_WMMA_F32_16X16X4_F32 : D = A(16x4) * B(4x16) + C(16x16), all f32
    return __builtin_amdgcn_wmma_f32_16x16x4_f32(
        /*neg_a=*/false, a, /*neg_b=*/false, b,
        /*c_mod=*/(short)0, c, /*reuse_a=*/false, /*reuse_b=*/false);
}

static __device__ __forceinline__ float rowmax16(float v) {
#pragma unroll
    for (int m = 1; m <= 8; m <<= 1) v = fmaxf(v, __shfl_xor(v, m, 32));
    return v;
}
static __device__ __forceinline__ float rowsum16(float v) {
#pragma unroll
    for (int m = 1; m <= 8; m <<= 1) v += __shfl_xor(v, m, 32);
    return v;
}

// Async copy 32B (global -> LDS) per lane, tracked by ASYNCcnt.
// IOFFSET applies to both the LDS and global addresses (CDNA5 ISA).
static __device__ __forceinline__ void async_copy_32B(void* lds, const void* g) {
    asm volatile(
        "global_load_async_to_lds_b128 %0, %1, off\n\t"
        "global_load_async_to_lds_b128 %0, %1, off offset:16"
        :: "v"((__attribute__((address_space(3))) void*)lds), "v"(g)
        : "memory");
}

static __device__ __forceinline__ void wait_async0() {
#if __has_builtin(__builtin_amdgcn_s_wait_asynccnt)
    __builtin_amdgcn_s_wait_asynccnt(0);
#else
    asm volatile("s_wait_asynccnt 0" ::: "memory");
#endif
}

// ---------------------------------------------------------------------------
// Y[l,n] = sum_c X[l,c] * W[n,c]   (X: Lq x Dm row-major, W: Dm x Dm row-major)
// SPLIT==1: write Y in split-head layout (H, Lq, DH); SPLIT==0: (Lq, Dm).
// Block: 256 threads (8 waves). Block tile: 64(M) x 64(N). K chunk = 32,
// double-buffered in LDS via async global->LDS copies.
// ---------------------------------------------------------------------------
template <int SPLIT>
__global__ __launch_bounds__(256) void gemm_xwt(const float* __restrict__ X,
                                                const float* __restrict__ W,
                                                float* __restrict__ Y) {
    __shared__ float Xs[2][64][36];  // +4 pad keeps 16B/8B alignment
    __shared__ float Ws[2][64][36];

    const int tid  = threadIdx.x;
    const int lane = tid & 31;
    const int wid  = tid >> 5;        // wave 0..7
    const int hi   = lane >> 4;       // half-wave select (K phase / M+8)
    const int ln   = lane & 15;
    const int m0   = blockIdx.x * 64;
    const int n0   = blockIdx.y * 64;
    const int mi   = wid & 3;         // tile row within block tile
    const int nj   = wid >> 2;        // tile cols nj and nj+2

    v8f c0 = {};
    v8f c1 = {};

    const int lr = tid >> 2;          // 0..63 : row staged by this thread
    const int lc = (tid & 3) * 8;     // 0,8,16,24 : 8 floats per thread
    const float* xrow = X + (size_t)(m0 + lr) * Dm + lc;
    const float* wrow = W + (size_t)(n0 + lr) * Dm + lc;

    // prologue: fill buffer 0 with chunk k0=0
    async_copy_32B(&Xs[0][lr][lc], xrow);
    async_copy_32B(&Ws[0][lr][lc], wrow);
    wait_async0();
    __syncthreads();

    for (int k0 = 0; k0 < Dm; k0 += 32) {
        const int buf = (k0 >> 5) & 1;
        if (k0 + 32 < Dm) {  // overlap next chunk's fill with this chunk's WMMAs
            async_copy_32B(&Xs[buf ^ 1][lr][lc], xrow + k0 + 32);
            async_copy_32B(&Ws[buf ^ 1][lr][lc], wrow + k0 + 32);
        }
#pragma unroll
        for (int kk = 0; kk < 32; kk += 4) {
            // A frag: lane<16 holds (M=ln, K=kk..kk+1); lane>=16 K=kk+2..kk+3
            v2f a  = *(const v2f*)&Xs[buf][mi * 16 + ln][kk + 2 * hi];
            // B frag: lane holds column n=ln of B = W rows (Y = X*W^T)
            v2f b0 = *(const v2f*)&Ws[buf][nj * 16 + ln][kk + 2 * hi];
            v2f b1 = *(const v2f*)&Ws[buf][(nj + 2) * 16 + ln][kk + 2 * hi];
            c0 = wmma_f32(a, b0, c0);
            c1 = wmma_f32(a, b1, c1);
        }
        wait_async0();
        __syncthreads();
    }

#pragma unroll
    for (int r = 0; r < 8; ++r) {
        const int l  = m0 + mi * 16 + r + 8 * hi;   // C layout: M = r + 8*hi
        const int nA = n0 + nj * 16 + ln;
        const int nB = n0 + (nj + 2) * 16 + ln;
        if (SPLIT) {
            Y[((size_t)(nA >> 6) * Lq + l) * DH + (nA & 63)] = c0[r];
            Y[((size_t)(nB >> 6) * Lq + l) * DH + (nB & 63)] = c1[r];
        } else {
            Y[(size_t)l * Dm + nA] = c0[r];
            Y[(size_t)l * Dm + nB] = c1[r];
        }
    }
}

// ---------------------------------------------------------------------------
// Flash attention over KV = concat(mem, projected). One wave per 16 query
// rows; 4 waves per block share double-buffered, async-staged K/V 16x64
// chunks (all waves in a block walk the same head & same KV sequence).
// Out: (Lq, Dm) merged-head attention output (pre-Wo), written to workspace.
// ---------------------------------------------------------------------------
__global__ __launch_bounds__(128) void flash_attn(
    const float* __restrict__ Q,    // (H, Lq, DH) workspace
    const float* __restrict__ Kp,   // (H, Lq, DH) projected k (in d_out)
    const float* __restrict__ Vp,   // (H, Lq, DH) projected v (in d_out)
    const float* __restrict__ MK,   // (H, Mm, DH) mem_k
    const float* __restrict__ MV,   // (H, Mm, DH) mem_v
    const unsigned char* __restrict__ mask,  // (Lq, KVN) bool
    float* __restrict__ Out) {
    __shared__ float Ks[2][16][72];
    __shared__ float Vs[2][16][72];
    __shared__ float Pl[4][16][18];  // per-wave P staging (C layout -> A layout)

    const int tid  = threadIdx.x;
    const int lane = tid & 31;
    const int wid  = tid >> 5;   // 0..3
    const int hi   = lane >> 4;
    const int ln   = lane & 15;
    const int h    = blockIdx.y;
    const int l0   = (blockIdx.x * 4 + wid) * 16;

    const int lr = tid >> 3;        // 0..15 KV row staged by this thread
    const int lc = (tid & 7) * 8;   // 8 floats per thread per matrix

    // Preload Q A-fragments: 16 steps of K=4 covering Dh=64
    v2f qa[16];
    {
        const float* qrow = Q + ((size_t)h * Lq + l0 + ln) * DH;
#pragma unroll
        for (int s = 0; s < 16; ++s)
            qa[s] = *(const v2f*)(qrow + 4 * s + 2 * hi);
    }

    float mrow[8], lsum[8];
#pragma unroll
    for (int r = 0; r < 8; ++r) { mrow[r] = -1.0e30f; lsum[r] = 0.0f; }
    v8f o[4] = {v8f{}, v8f{}, v8f{}, v8f{}};

    // prologue: async-fill buffer 0 with chunk n0=0 (lr < 16 <= Mm -> mem side)
    async_copy_32B(&Ks[0][lr][lc], MK + ((size_t)h * Mm + lr) * DH + lc);
    async_copy_32B(&Vs[0][lr][lc], MV + ((size_t)h * Mm + lr) * DH + lc);
    wait_async0();
    __syncthreads();

    for (int n0 = 0; n0 < KVN; n0 += 16) {
        const int buf = (n0 >> 4) & 1;

        // overlap: issue async fill of the NEXT chunk into the other buffer
        if (n0 + 16 < KVN) {
            const int kv = n0 + 16 + lr;
            const float* krp = (kv < Mm) ? (MK + ((size_t)h * Mm + kv) * DH)
                                         : (Kp + ((size_t)h * Lq + (kv - Mm)) * DH);
            const float* vrp = (kv < Mm) ? (MV + ((size_t)h * Mm + kv) * DH)
                                         : (Vp + ((size_t)h * Lq + (kv - Mm)) * DH);
            async_copy_32B(&Ks[buf ^ 1][lr][lc], krp + lc);
            async_copy_32B(&Vs[buf ^ 1][lr][lc], vrp + lc);
        }

        // S = Q * K^T  (16q x 16kv), K dim = Dh = 64 -> 16 WMMAs
        v8f s = {};
#pragma unroll
        for (int ss = 0; ss < 16; ++ss) {
            v2f b = *(const v2f*)&Ks[buf][ln][4 * ss + 2 * hi];  // B[d][n]=Kcat[n0+n][d]
            s = wmma_f32(qa[ss], b, s);
        }

        // scale 1/sqrt(Dh), mask, online softmax; stage P into LDS
#pragma unroll
        for (int r = 0; r < 8; ++r) {
            const int l = l0 + r + 8 * hi;                       // C layout row
            const float keep = mask[(size_t)l * KVN + n0 + ln] ? 1.0f : 0.0f;
            float sv = s[r] * 0.125f;
            sv = keep != 0.0f ? sv : -1.0e30f;
            const float mx    = rowmax16(sv);
            const float mnew  = fmaxf(mrow[r], mx);
            const float alpha = __expf(mrow[r] - mnew);
            mrow[r] = mnew;
            const float p = __expf(sv - mnew) * keep;
            lsum[r] = alpha * lsum[r] + rowsum16(p);
#pragma unroll
            for (int t = 0; t < 4; ++t) o[t][r] *= alpha;
            Pl[wid][r + 8 * hi][ln] = p;
        }
        __syncthreads();  // P visible (uniform across block)

        // O += P(16x16) * Vcat(16x64): 4 Dh tiles x 4 K-steps
#pragma unroll
        for (int t = 0; t < 4; ++t) {
#pragma unroll
            for (int ss = 0; ss < 4; ++ss) {
                v2f a = *(const v2f*)&Pl[wid][ln][4 * ss + 2 * hi];  // A frag of P
                v2f b;
                b.x = Vs[buf][4 * ss + 2 * hi][16 * t + ln];   // B[k][n]=V[n0+k][16t+n]
                b.y = Vs[buf][4 * ss + 2 * hi + 1][16 * t + ln];
                o[t] = wmma_f32(a, b, o[t]);
            }
        }

        wait_async0();     // next chunk's LDS fill complete (this wave's issues)
        __syncthreads();   // ... and everyone else's; also guards Pl reuse
    }

    // normalize and write merged-head output (Lq, Dm)
#pragma unroll
    for (int r = 0; r < 8; ++r) {
        const float inv = 1.0f / lsum[r];  // last kv slot always unmasked -> lsum>0
        const int l = l0 + r + 8 * hi;
#pragma unroll
        for (int t = 0; t < 4; ++t)
            Out[(size_t)l * Dm + h * DH + 16 * t + ln] = o[t][r] * inv;
    }
}

// ---------------------------------------------------------------------------
extern "C" void kernel_launch(void* const* d_in, const int* in_sizes, int n_in,
                              void* d_out, int out_size, void* d_ws, size_t ws_size,
                              hipStream_t stream) {
    const float*         x     = (const float*)d_in[0];
    const float*         mem_k = (const float*)d_in[1];
    const float*         mem_v = (const float*)d_in[2];
    const unsigned char* mask  = (const unsigned char*)d_in[3];
    const float*         Wq    = (const float*)d_in[4];
    const float*         Wk    = (const float*)d_in[5];
    const float*         Wv    = (const float*)d_in[6];
    const float*         Wo    = (const float*)d_in[7];

    float* y     = (float*)d_out;                       // (Lq, Dm)
    float* k_out = y + (size_t)Lq * Dm;                 // (H, Lq, DH)
    float* v_out = k_out + (size_t)Hh * Lq * DH;        // (H, Lq, DH)

    float* q_ws   = (float*)d_ws;                       // (H, Lq, DH) = 8 MB
    float* att_ws = q_ws + (size_t)Hh * Lq * DH;        // (Lq, Dm)    = 8 MB

    dim3 gG(Lq / 64, Dm / 64), bG(256);
    gemm_xwt<1><<<gG, bG, 0, stream>>>(x, Wq, q_ws);
    gemm_xwt<1><<<gG, bG, 0, stream>>>(x, Wk, k_out);
    gemm_xwt<1><<<gG, bG, 0, stream>>>(x, Wv, v_out);

    dim3 gA(Lq / 64, Hh), bA(128);
    flash_attn<<<gA, bA, 0, stream>>>(q_ws, k_out, v_out, mem_k, mem_v, mask, att_ws);

    gemm_xwt<0><<<gG, bG, 0, stream>>>(att_ws, Wo, y);
}